// CP1_17669495456474
// MI455X (gfx1250) — compile-verified
//
#include <hip/hip_runtime.h>
#include <cmath>

// ---------------------------------------------------------------------------
// Contextual-attention pipeline for MI455X (gfx1250, wave32).
// B=4, C=64, H=W=64. L=1024 bkg patches (stride2), P=63*63 fg positions,
// K = C*4*4 = 1024.  Core op: per-batch 1024x3969x1024 fp32 GEMM done with
// V_WMMA_F32_16X16X4_F32 (fp32 preserved end-to-end).
// N padded to 4032 so diag3 over flattened indices can read exact zeros.
// ---------------------------------------------------------------------------

typedef float v2f __attribute__((ext_vector_type(2)));
typedef float v8f __attribute__((ext_vector_type(8)));

#define BATCH 4
#define CCH   64
#define HH    64
#define WW    64
#define LL    1024   // 32*32 background patches
#define KK    1024   // C * 4 * 4
#define HS    63
#define PP    3969   // 63*63
#define NP    4032   // padded N (63 * 64)

__device__ __forceinline__ int clamp63(int v) { return v < 0 ? 0 : (v > 63 ? 63 : v); }

// ---- per-(b,c) inverse L2 norm of b over HxW ------------------------------
__global__ void k_norms(const float* __restrict__ bsrc, float* __restrict__ invn) {
    __shared__ float red[256];
    int bc = blockIdx.x;                       // 0..255
    const float* src = bsrc + (size_t)bc * (HH * WW);
    float s = 0.f;
    for (int i = threadIdx.x; i < HH * WW; i += 256) { float v = src[i]; s += v * v; }
    red[threadIdx.x] = s;
    __syncthreads();
    for (int off = 128; off > 0; off >>= 1) {
        if (threadIdx.x < (unsigned)off) red[threadIdx.x] += red[threadIdx.x + off];
        __syncthreads();
    }
    if (threadIdx.x == 0) invn[bc] = 1.0f / sqrtf(red[0] + 1e-8f);
}

// ---- im2col of normalized b -> A[b][l][k], row-major, K=1024 --------------
__global__ void k_packA(const float* __restrict__ bsrc, const float* __restrict__ invn,
                        float* __restrict__ A) {
    int bl = blockIdx.x;                       // b*1024 + l
    int b = bl >> 10, l = bl & 1023;
    int lh = l >> 5, lw = l & 31;
    float* dst = A + (size_t)bl * KK;
    for (int k = threadIdx.x; k < KK; k += 256) {
        int c = k >> 4, ij = k & 15, i = ij >> 2, j = ij & 3;
        int y = clamp63(2 * lh + i - 1);       // edge padding
        int x = clamp63(2 * lw + j - 1);
        int bc = (b << 6) + c;
        dst[k] = bsrc[((size_t)bc * 64 + y) * 64 + x] * invn[bc];
    }
}

// ---- im2col of f -> B[b][k][n], n padded to 4032 with zeros ---------------
__global__ void k_packB(const float* __restrict__ f, float* __restrict__ Bm) {
    int bk = blockIdx.x;                       // b*1024 + k
    int b = bk >> 10, k = bk & 1023;
    int c = k >> 4, ij = k & 15, i = ij >> 2, j = ij & 3;
    const float* src = f + (size_t)((b << 6) + c) * (HH * WW);
    float* dst = Bm + (size_t)bk * NP;
    for (int n = threadIdx.x; n < NP; n += 256) {
        float v = 0.f;
        if (n < PP) {
            int y = n / 63, x = n - y * 63;
            v = src[clamp63(y + i - 1) * 64 + clamp63(x + j - 1)];
        }
        dst[n] = v;
    }
}

// ---- patch means of (1-mask): mmk (stride2 grid), mmp (stride1 grid) ------
__global__ void k_mmk(const float* __restrict__ mask, float* __restrict__ mmk) {
    int idx = blockIdx.x * 256 + threadIdx.x;
    if (idx >= BATCH * LL) return;
    int b = idx >> 10, l = idx & 1023;
    int lh = l >> 5, lw = l & 31;
    const float* m = mask + (size_t)b * HH * WW;
    float s = 0.f;
    for (int i = 0; i < 4; ++i)
        for (int j = 0; j < 4; ++j)
            s += 1.0f - m[clamp63(2 * lh + i - 1) * 64 + clamp63(2 * lw + j - 1)];
    mmk[idx] = s * (1.0f / 16.0f);
}

__global__ void k_mmp(const float* __restrict__ mask, float* __restrict__ mmp) {
    int idx = blockIdx.x * 256 + threadIdx.x;
    if (idx >= BATCH * PP) return;
    int b = idx / PP, p = idx - b * PP;
    int py = p / 63, px = p - py * 63;
    const float* m = mask + (size_t)b * HH * WW;
    float s = 0.f;
    for (int i = 0; i < 4; ++i)
        for (int j = 0; j < 4; ++j)
            s += 1.0f - m[clamp63(py + i - 1) * 64 + clamp63(px + j - 1)];
    mmp[idx] = s * (1.0f / 16.0f);
}

// ---- fp32 WMMA GEMM: C[b] = A[b] (LLxKK) * B[b] (KKxNP) -------------------
// 256 thr = 8 waves; wave -> 32x32 C tile (2x2 WMMA 16x16), block -> 128x64.
__global__ __launch_bounds__(256) void k_gemm(const float* __restrict__ A,
                                              const float* __restrict__ Bm,
                                              float* __restrict__ Cc) {
    const int lane = threadIdx.x & 31;
    const int wave = threadIdx.x >> 5;
    const int wm = wave >> 1;                  // 0..3
    const int wn = wave & 1;                   // 0..1
    const int b  = blockIdx.z;
    const int m0 = blockIdx.y * 128 + wm * 32;
    const int n0 = blockIdx.x * 64 + wn * 32;
    const int l15 = lane & 15;
    const int hi  = lane >> 4;                 // 0: K pair {0,1}, 1: K pair {2,3}

    const float* Ab = A  + (size_t)b * LL * KK;
    const float* Bb = Bm + (size_t)b * KK * NP;
    float*       Cb = Cc + (size_t)b * LL * NP;

    v8f c00 = {}, c01 = {}, c10 = {}, c11 = {};

    // A 16x4 frag: lane(0-15)=M, VGPR0/1 = K {hi*2, hi*2+1}  -> v2f load
    const float* pa0 = Ab + (size_t)(m0 + l15) * KK + hi * 2;
    const float* pa1 = Ab + (size_t)(m0 + 16 + l15) * KK + hi * 2;
    // B 4x16 frag: lane(0-15)=N, VGPR0/1 = K {hi*2, hi*2+1}
    const float* pb0 = Bb + (size_t)(hi * 2) * NP + n0 + l15;
    const float* pb1 = pb0 + 16;

    for (int k = 0; k < KK; k += 4) {
        v2f a0 = *(const v2f*)pa0; pa0 += 4;
        v2f a1 = *(const v2f*)pa1; pa1 += 4;
        v2f b0; b0.x = pb0[0]; b0.y = pb0[NP];
        v2f b1; b1.x = pb1[0]; b1.y = pb1[NP];
        pb0 += 4 * (size_t)NP; pb1 += 4 * (size_t)NP;
        c00 = __builtin_amdgcn_wmma_f32_16x16x4_f32(false, a0, false, b0, (short)0, c00, false, false);
        c01 = __builtin_amdgcn_wmma_f32_16x16x4_f32(false, a0, false, b1, (short)0, c01, false, false);
        c10 = __builtin_amdgcn_wmma_f32_16x16x4_f32(false, a1, false, b0, (short)0, c10, false, false);
        c11 = __builtin_amdgcn_wmma_f32_16x16x4_f32(false, a1, false, b1, (short)0, c11, false, false);
    }

    // C/D layout: VGPR r -> M = r (+8 for lanes 16-31), N = lane&15
    #pragma unroll
    for (int r = 0; r < 8; ++r) {
        int row0 = m0 + r + hi * 8;
        int row1 = m0 + 16 + r + hi * 8;
        int col  = n0 + l15;
        Cb[(size_t)row0 * NP + col]      = c00[r];
        Cb[(size_t)row0 * NP + col + 16] = c01[r];
        Cb[(size_t)row1 * NP + col]      = c10[r];
        Cb[(size_t)row1 * NP + col + 16] = c11[r];
    }
}

// ---- pass 1: diag3 over flattened (r = L index, c = spatial index) --------
__global__ void k_fuse1(const float* __restrict__ Cc, float* __restrict__ T1) {
    int bl = blockIdx.x;
    int b = bl >> 10, r = bl & 1023;
    const float* base = Cc + (size_t)b * LL * NP;
    const float* s0 = base + (size_t)r * NP;
    const float* sm = base + (size_t)(r - 1) * NP;
    const float* sp = base + (size_t)(r + 1) * NP;
    float* dst = T1 + (size_t)bl * NP;
    bool hm = r > 0, hp = r < LL - 1;
    for (int c = threadIdx.x; c < PP; c += 256) {
        float v = s0[c];
        if (hm && c > 0) v += sm[c - 1];
        if (hp)          v += sp[c + 1];   // c+1==3969 hits the zero pad column
        dst[c] = v;
    }
}

// ---- pass 2: diag3 in swapped index space, mapped back --------------------
__global__ void k_fuse2(const float* __restrict__ T1, float* __restrict__ Fo) {
    int bl = blockIdx.x;
    int b = bl >> 10, l = bl & 1023;
    int lh = l >> 5, lw = l & 31;
    int rp = lw * 32 + lh;                     // primed row; center maps back to l
    const float* src = T1 + (size_t)b * LL * NP;
    float* dst = Fo + (size_t)bl * NP;
    int rA = -1, rB = -1;
    if (rp > 0)      { int q = rp - 1; rA = (q & 31) * 32 + (q >> 5); }
    if (rp < LL - 1) { int q = rp + 1; rB = (q & 31) * 32 + (q >> 5); }
    for (int p = threadIdx.x; p < PP; p += 256) {
        int x = p % 63, y = p / 63;
        int cp = x * 63 + y;                   // primed col
        float v = src[(size_t)l * NP + p];
        if (rA >= 0 && cp > 0) {
            int q = cp - 1;
            v += src[(size_t)rA * NP + (q % 63) * 63 + q / 63];
        }
        if (rB >= 0 && cp < PP - 1) {
            int q = cp + 1;
            v += src[(size_t)rB * NP + (q % 63) * 63 + q / 63];
        }
        dst[p] = v;
    }
}

// ---- mask + scaled softmax over L, online max/sum -------------------------
__global__ __launch_bounds__(256) void k_softmax(const float* __restrict__ Fo,
                                                 const float* __restrict__ mmk,
                                                 const float* __restrict__ mmp,
                                                 float* __restrict__ out) {
    __shared__ float smk[LL];
    int b = blockIdx.y;
    int p = blockIdx.x * 256 + threadIdx.x;
    for (int i = threadIdx.x; i < LL; i += 256) smk[i] = mmk[b * LL + i];
    __syncthreads();
    if (p >= PP) return;
    float mp = mmp[(size_t)b * PP + p];
    bool ppp = mp > 0.5f;
    const float* col = Fo + (size_t)b * LL * NP + p;
    float mx = -3.0e38f, sum = 0.f;
    for (int r = 0; r < LL; ++r) {
        float mk = smk[r];
        bool mm = (ppp && (mk > mp)) || (mk == 1.0f);
        float z = mm ? col[(size_t)r * NP] * 10.0f : 0.0f;
        float nm = fmaxf(mx, z);
        sum = sum * __expf(mx - nm) + __expf(z - nm);
        mx = nm;
    }
    float inv = 1.0f / sum;
    float* o = out + (size_t)b * LL * PP + p;
    for (int r = 0; r < LL; ++r) {
        float mk = smk[r];
        bool mm = (ppp && (mk > mp)) || (mk == 1.0f);
        float z = mm ? col[(size_t)r * NP] * 10.0f : 0.0f;
        o[(size_t)r * PP] = __expf(z - mx) * inv;
    }
}

// ---------------------------------------------------------------------------
extern "C" void kernel_launch(void* const* d_in, const int* in_sizes, int n_in,
                              void* d_out, int out_size, void* d_ws, size_t ws_size,
                              hipStream_t stream) {
    const float* f    = (const float*)d_in[0];
    const float* bsrc = (const float*)d_in[1];
    const float* mask = (const float*)d_in[2];
    float* out = (float*)d_out;

    // workspace layout (floats); ~150 MB total, buffers reused across phases
    float* ws   = (float*)d_ws;
    float* invn = ws;                          // 256
    float* mmk  = ws + 256;                    // 4096
    float* mmp  = ws + 256 + 4096;             // 15876
    float* A    = ws + 32768;                                  // 4*1024*1024
    float* Bm   = A + (size_t)BATCH * LL * KK;                 // 4*1024*4032
    float* Cc   = Bm + (size_t)BATCH * KK * NP;                // 4*1024*4032

    k_norms<<<BATCH * CCH, 256, 0, stream>>>(bsrc, invn);
    k_packA<<<BATCH * LL, 256, 0, stream>>>(bsrc, invn, A);
    k_packB<<<BATCH * KK, 256, 0, stream>>>(f, Bm);
    k_mmk<<<(BATCH * LL + 255) / 256, 256, 0, stream>>>(mask, mmk);
    k_mmp<<<(BATCH * PP + 255) / 256, 256, 0, stream>>>(mask, mmp);
    k_gemm<<<dim3(NP / 64, LL / 128, BATCH), 256, 0, stream>>>(A, Bm, Cc);
    k_fuse1<<<BATCH * LL, 256, 0, stream>>>(Cc, Bm);   // T1 overwrites Bm (GEMM done)
    k_fuse2<<<BATCH * LL, 256, 0, stream>>>(Bm, Cc);   // fused overwrites Cc
    k_softmax<<<dim3((PP + 255) / 256, BATCH), 256, 0, stream>>>(Cc, mmk, mmp, out);
}